// TrainableMetaModel_27230092657385
// MI455X (gfx1250) — compile-verified
//
#include <hip/hip_runtime.h>
#include <cstdint>

#define MP      2048
#define NC      4
#define TSTEPS  128
#define BETA_   0.25f
#define NBLK    64
#define NTHR    256
#define RPB     (MP / NBLK)                       /* 32 rows per block  */
#define OUT_ELEMS ((size_t)TSTEPS * MP * (NC + 1))/* 1310720            */

typedef float v2f __attribute__((ext_vector_type(2)));
typedef float v8f __attribute__((ext_vector_type(8)));

struct Ws {
  unsigned counter;
  unsigned gen;
  unsigned pad[2];
  float ntot[MP];        // column sums of R
  float p[MP];           // infection probabilities (shared across blocks)
  float c0[2][MP];       // compartment-0 vector, double buffered (shared)
  float rho[2][MP * NC]; // full state, double buffered (block-private by row)
  float partials[NBLK];  // agreement partial sums
};

// ---------------- device-wide sense-reversal barrier ----------------
__device__ __forceinline__ void grid_barrier(Ws* ws) {
  __syncthreads();
  __threadfence();  // release: our global stores reach agent scope
  if (threadIdx.x == 0) {
    unsigned g = __hip_atomic_load(&ws->gen, __ATOMIC_RELAXED, __HIP_MEMORY_SCOPE_AGENT);
    unsigned a = __hip_atomic_fetch_add(&ws->counter, 1u, __ATOMIC_ACQ_REL, __HIP_MEMORY_SCOPE_AGENT);
    if (a == (unsigned)(gridDim.x - 1)) {
      __hip_atomic_store(&ws->counter, 0u, __ATOMIC_RELAXED, __HIP_MEMORY_SCOPE_AGENT);
      __hip_atomic_store(&ws->gen, g + 1u, __ATOMIC_RELEASE, __HIP_MEMORY_SCOPE_AGENT);
    } else {
      while (__hip_atomic_load(&ws->gen, __ATOMIC_ACQUIRE, __HIP_MEMORY_SCOPE_AGENT) == g) {
        __builtin_amdgcn_s_sleep(1);
      }
    }
  }
  __syncthreads();
  __threadfence();  // acquire: invalidate stale WGP$ lines before reading
}

// ------------- async-stage 8KB vector from global into LDS ----------
__device__ __forceinline__ void stage_lds(float* xs, const float* __restrict__ src) {
  __syncthreads();  // previous consumers of xs are done
  unsigned lofs = (unsigned)(uintptr_t)(xs + threadIdx.x * 8);
  uint64_t gaddr = (uint64_t)(uintptr_t)(src + threadIdx.x * 8);
  // CDNA5 async LDS-load path (ASYNCcnt-tracked): each lane moves 32 B.
  asm volatile("global_load_async_to_lds_b128 %0, %1, off\n\t"
               "global_load_async_to_lds_b128 %0, %1, off offset:16"
               :: "v"(lofs), "v"(gaddr) : "memory");
  asm volatile("s_wait_asynccnt 0" ::: "memory");
  __syncthreads();
}

// ------------- 4-row dot products for one wave (fp32, b128) ---------
__device__ __forceinline__ void matvec_rows4(const float* __restrict__ R,
                                             const float* xs, int base_row,
                                             float acc[4]) {
  const int lane = threadIdx.x & 31;
  acc[0] = acc[1] = acc[2] = acc[3] = 0.f;
  for (int j = 0; j < MP; j += 128) {
    const int col = j + lane * 4;
    const float4 xv = *(const float4*)(xs + col);   // ds_load_b128
#pragma unroll
    for (int r = 0; r < 4; ++r) {
      const float4 rv = *(const float4*)(R + (size_t)(base_row + r) * MP + col);
      acc[r] = fmaf(rv.x, xv.x, acc[r]);
      acc[r] = fmaf(rv.y, xv.y, acc[r]);
      acc[r] = fmaf(rv.z, xv.z, acc[r]);
      acc[r] = fmaf(rv.w, xv.w, acc[r]);
    }
  }
#pragma unroll
  for (int r = 0; r < 4; ++r) {
    float v = acc[r];
#pragma unroll
    for (int off = 16; off > 0; off >>= 1) v += __shfl_xor(v, off, 32);
    acc[r] = v;  // full row sum in every lane
  }
}

// --------------------------- init kernel ----------------------------
__global__ void __launch_bounds__(NTHR) tm_init_kernel(const float* __restrict__ rho0, Ws* ws) {
  if (threadIdx.x == 0) { ws->counter = 0u; ws->gen = 0u; }
  for (int i = threadIdx.x; i < MP * NC; i += NTHR) ws->rho[0][i] = rho0[i];
  for (int i = threadIdx.x; i < MP; i += NTHR) ws->c0[0][i] = rho0[i * NC];
}

// ----------------------- persistent main kernel ---------------------
__global__ void __launch_bounds__(NTHR) tm_meta_kernel(const float* __restrict__ R,
                                                       const float* __restrict__ Tm,
                                                       Ws* ws, float* __restrict__ out) {
  __shared__ __align__(16) float xs[MP];        // staged shared vector (8 KB)
  __shared__ float lds_ni[RPB];                 // new_infected (block rows)
  __shared__ __align__(16) float lds_rho[RPB * 16];  // rho' tile, stride 16 (2 KB)
  __shared__ float rbuf[NTHR];

  const int tid  = threadIdx.x;
  const int blk  = blockIdx.x;
  const int wave = tid >> 5;
  const int lane = tid & 31;
  const int base_row = blk * RPB + wave * 4;

  // ---- prologue: ntot[j] = sum_i R[i][j] (blocks 0..7, coalesced) ----
  if (blk < MP / NTHR) {
    const int col = blk * NTHR + tid;
    float s = 0.f;
    for (int i = 0; i < MP; ++i) s += R[(size_t)i * MP + col];
    ws->ntot[col] = s;
  }
  grid_barrier(ws);

  // ---- loop-invariant WMMA B-operand: T zero-padded to 4x16 ----
  // ISA f32 B layout: lane n (0-15) VGPR0=B[0,n], VGPR1=B[1,n];
  //                   lane n+16      VGPR0=B[2,n], VGPR1=B[3,n].
  const int half = lane >> 4;
  const int mn   = lane & 15;
  v2f bmat;
  {
    const int   cn  = (mn < NC) ? mn : 0;   // clamped index: no predication
    const float msk = (mn < NC) ? 1.f : 0.f;
    bmat.x = Tm[(half * 2 + 0) * NC + cn] * msk;
    bmat.y = Tm[(half * 2 + 1) * NC + cn] * msk;
  }
  const float nim = (mn == 0) ? 1.f : 0.f;  // mask for C-operand column 0
  // ---- loop-invariant ntot for this wave's 4 rows ----
  float ntot_r[4];
#pragma unroll
  for (int r = 0; r < 4; ++r) ntot_r[r] = ws->ntot[base_row + r];

  for (int t = 0; t < TSTEPS; ++t) {
    const int cb = t & 1, nb = cb ^ 1;
    const float* rho_c = ws->rho[cb];
    float*       rho_n = ws->rho[nb];
    float acc[4];

    // ---- phase A: v = R @ c0 ; p = 1 - exp(-beta * v / ntot) ----
    stage_lds(xs, ws->c0[cb]);
    matvec_rows4(R, xs, base_row, acc);
    if (lane == 0) {
#pragma unroll
      for (int r = 0; r < 4; ++r) {
        ws->p[base_row + r] = 1.f - expf(-BETA_ * acc[r] / ntot_r[r]);
      }
    }
    grid_barrier(ws);  // p must be complete device-wide

    // ---- phase B: w = R @ p ; new_infected = (1 - sum(rho)) * w ----
    stage_lds(xs, ws->p);
    matvec_rows4(R, xs, base_row, acc);
    if (lane == 0) {
#pragma unroll
      for (int r = 0; r < 4; ++r) {
        const float4 rc = *(const float4*)(rho_c + (size_t)(base_row + r) * NC);
        const float ssum = rc.x + rc.y + rc.z + rc.w;
        lds_ni[wave * 4 + r] = (1.f - ssum) * acc[r];
      }
    }
    __syncthreads();

    // ---- phase C: rho' = rho @ T + ni*e0 as ONE v_wmma_f32_16x16x4_f32 ----
    // A = 16x4 rho tile, B = T (hoisted), C[m,0] = new_infected[m] (rest 0).
    // Fully branchless: masks are arithmetic, EXEC is the uniform wave mask.
    if (wave < 2) {
      const int tileb = blk * RPB + wave * 16;
      v2f a;
      const float* ar = rho_c + (size_t)(tileb + mn) * NC + half * 2;
      a.x = ar[0];
      a.y = ar[1];
      v8f c;
#pragma unroll
      for (int j = 0; j < 8; ++j) {      // broadcast ds_loads, no predication
        c[j] = nim * lds_ni[wave * 16 + half * 8 + j];
      }
      c = __builtin_amdgcn_wmma_f32_16x16x4_f32(false, a, false, bmat, (short)0,
                                                c, false, false);
#pragma unroll
      for (int j = 0; j < 8; ++j) {      // D: VGPR j holds M=j / M=j+8
        const int lr = wave * 16 + half * 8 + j;
        lds_rho[lr * 16 + mn] = fminf(fmaxf(c[j], 0.f), 1e10f);
      }
    }
    __syncthreads();

    // ---- coalesced writeback: rho row, c0, S + traj row ----
    if (tid < RPB) {
      const int row = blk * RPB + tid;
      const float4 rn = *(const float4*)(lds_rho + tid * 16);  // cols 0..3
      *(float4*)(rho_n + (size_t)row * NC) = rn;     // b128 store
      ws->c0[nb][row] = rn.x;                        // contiguous c0 for phase A
      float* ob = out + ((size_t)t * MP + row) * (NC + 1);
      ob[0] = 1.f - (rn.x + rn.y + rn.z + rn.w);     // S
      ob[1] = rn.x; ob[2] = rn.y; ob[3] = rn.z; ob[4] = rn.w;
    }
    grid_barrier(ws);  // rho'/c0' complete device-wide before next step
  }

  // ---- agreement = mean(|traj|), deterministic two-level reduction ----
  const size_t chunk = OUT_ELEMS / NBLK;  // 20480
  float local = 0.f;
  for (size_t i = tid; i < chunk; i += NTHR) local += fabsf(out[(size_t)blk * chunk + i]);
  rbuf[tid] = local;
  __syncthreads();
  for (int s = NTHR / 2; s > 0; s >>= 1) {
    if (tid < s) rbuf[tid] += rbuf[tid + s];
    __syncthreads();
  }
  if (tid == 0) ws->partials[blk] = rbuf[0];
  grid_barrier(ws);
  if (blk == 0 && tid == 0) {
    float tot = 0.f;
    for (int i = 0; i < NBLK; ++i) tot += ws->partials[i];
    out[OUT_ELEMS] = tot * (1.f / (float)OUT_ELEMS);
  }
}

extern "C" void kernel_launch(void* const* d_in, const int* in_sizes, int n_in,
                              void* d_out, int out_size, void* d_ws, size_t ws_size,
                              hipStream_t stream) {
  (void)in_sizes; (void)n_in; (void)out_size; (void)ws_size;
  const float* R    = (const float*)d_in[0];  // (1, 2048, 2048)
  const float* Tm   = (const float*)d_in[1];  // (1, 4, 4)
  const float* rho0 = (const float*)d_in[2];  // (1, 2048, 4)
  float* out = (float*)d_out;                 // traj (1310720) + agreement (1)
  Ws* ws = (Ws*)d_ws;

  tm_init_kernel<<<1, NTHR, 0, stream>>>(rho0, ws);
  tm_meta_kernel<<<NBLK, NTHR, 0, stream>>>(R, Tm, ws, out);
}